// FPModule_26353919328302
// MI455X (gfx1250) — compile-verified
//
#include <hip/hip_runtime.h>
#include <hip/hip_bf16.h>

typedef __attribute__((ext_vector_type(2))) float v2f;
typedef __attribute__((ext_vector_type(8))) float v8f;

// ---- Optional CDNA5 async memory->LDS path (guarded: falls back cleanly) ----
#if __has_builtin(__builtin_amdgcn_global_load_async_to_lds_b128)
#define HAVE_ASYNC_LDS 1
typedef int vint4 __attribute__((vector_size(16)));
#define AS1(p) ((__attribute__((address_space(1))) vint4*)(p))
#define AS3(p) ((__attribute__((address_space(3))) vint4*)(p))
#if __has_builtin(__builtin_amdgcn_s_wait_asynccnt)
#define WAIT_ASYNC() __builtin_amdgcn_s_wait_asynccnt(0)
#else
#define WAIT_ASYNC() asm volatile("s_wait_asynccnt 0" ::: "memory")
#endif
#endif

// ---------------------------------------------------------------------------
// 1) kNN top-3: one thread per target point, source positions staged in LDS.
// ---------------------------------------------------------------------------
__launch_bounds__(256)
__global__ void knn_kernel(const float* __restrict__ pos,
                           const float* __restrict__ pos_skip,
                           int* __restrict__ idxOut,
                           float* __restrict__ wOut)
{
    __shared__ float sx[2048], sy[2048], sz[2048];
    const int tid = threadIdx.x;
    const int b = blockIdx.x >> 5;                 // 32 blocks per batch
    const int mLocal = ((blockIdx.x & 31) << 8) + tid;
    const int srcBase = b * 2048;

    for (int p = tid; p < 2048; p += 256) {
        const float* pp = &pos[(size_t)(srcBase + p) * 3];
        sx[p] = pp[0]; sy[p] = pp[1]; sz[p] = pp[2];
    }
    __syncthreads();

    const int tgt = b * 8192 + mLocal;
    const float tx = pos_skip[(size_t)tgt * 3 + 0];
    const float ty = pos_skip[(size_t)tgt * 3 + 1];
    const float tz = pos_skip[(size_t)tgt * 3 + 2];

    float d0 = 3.4e38f, d1 = 3.4e38f, d2 = 3.4e38f;
    int i0 = 0, i1 = 0, i2 = 0;
    for (int n = 0; n < 2048; ++n) {
        const float dx = tx - sx[n];
        const float dy = ty - sy[n];
        const float dz = tz - sz[n];
        const float d = dx * dx + dy * dy + dz * dz;
        if (d < d2) {
            if (d < d1) {
                d2 = d1; i2 = i1;
                if (d < d0) { d1 = d0; i1 = i0; d0 = d; i0 = n; }
                else        { d1 = d;  i1 = n; }
            } else { d2 = d; i2 = n; }
        }
    }
    const float w0 = 1.f / fmaxf(d0, 1e-16f);
    const float w1 = 1.f / fmaxf(d1, 1e-16f);
    const float w2 = 1.f / fmaxf(d2, 1e-16f);
    const float inv = 1.f / (w0 + w1 + w2);
    idxOut[tgt * 3 + 0] = srcBase + i0;
    idxOut[tgt * 3 + 1] = srcBase + i1;
    idxOut[tgt * 3 + 2] = srcBase + i2;
    wOut[tgt * 3 + 0] = w0 * inv;
    wOut[tgt * 3 + 1] = w1 * inv;
    wOut[tgt * 3 + 2] = w2 * inv;
}

// ---------------------------------------------------------------------------
// 2) Weighted feature gather: Y[m, :] = sum_k w_k * x[idx_k, :]   (L2-resident x)
// ---------------------------------------------------------------------------
__launch_bounds__(128)
__global__ void interp_kernel(const float* __restrict__ x,
                              const int* __restrict__ idx,
                              const float* __restrict__ w,
                              float* __restrict__ Y)
{
    const int m = blockIdx.x;
    const int i0 = idx[m * 3 + 0], i1 = idx[m * 3 + 1], i2 = idx[m * 3 + 2];
    const float w0 = w[m * 3 + 0], w1 = w[m * 3 + 1], w2 = w[m * 3 + 2];
    for (int c = threadIdx.x; c < 256; c += 128) {
        Y[(size_t)m * 256 + c] = w0 * x[(size_t)i0 * 256 + c]
                               + w1 * x[(size_t)i1 * 256 + c]
                               + w2 * x[(size_t)i2 * 256 + c];
    }
}

// ---------------------------------------------------------------------------
// 3) Split-K fused GEMM: Out = relu([A | A2] @ W + bias), optional fused
//    batchnorm applied to A elements on load (a*scale[k] + shift[k]).
//    f32 WMMA 16x16x4; block = 8 waves -> 128x64 tile; wave -> 32x32
//    (4 accumulators; A- and B-frags each reused twice).
//    Raw-copy tiles staged via GLOBAL_LOAD_ASYNC_TO_LDS_B128 when available.
// ---------------------------------------------------------------------------
__launch_bounds__(256)
__global__ void gemm_relu_kernel(const float* __restrict__ A, int ldA,
                                 const float* __restrict__ A2, int ldA2,
                                 int Ksplit, int K,
                                 const float* __restrict__ W,
                                 const float* __restrict__ bias,
                                 const float* __restrict__ bnScale,
                                 const float* __restrict__ bnShift,
                                 float* __restrict__ Out)
{
    // Stride 20: b64 A-frag reads -> all 32 lanes hit disjoint bank pairs
    // (20m mod 64 is always 0 mod 4; the +2-float half-wave offset cannot alias).
    __shared__ float As[128][20];
    // Stride 72: half-wave row-pair offset = 576B = 16 banks, disjoint from the
    // 16 consecutive-column banks; float4/async-b128 stores stay 16B-aligned.
    __shared__ float Bs[16][72];

    const int tid   = threadIdx.x;
    const int lane  = tid & 31;
    const int wave  = tid >> 5;
    const int waveM = wave >> 1;    // 0..3  -> 32-row strip
    const int waveN = wave & 1;     // 0..1  -> 32-col strip
    const int hi    = lane >> 4;    // half-wave select (K+2 / M+8)
    const int lr    = lane & 15;
    const int rowBase = blockIdx.x * 128;
    const int colBase = blockIdx.y * 64;

    v8f acc00 = {}, acc01 = {}, acc10 = {}, acc11 = {};
    const int colW0 = colBase + waveN * 32 + lr;
    const float bias0 = bias[colW0];
    const float bias1 = bias[colW0 + 16];

    for (int kb = 0; kb < K; kb += 16) {
        // Pick A source for this 16-wide K-block (Ksplit is 16-aligned).
        const float* src; int ldS; int kk;
        if (kb < Ksplit) { src = A;  ldS = ldA;  kk = kb; }
        else             { src = A2; ldS = ldA2; kk = kb - Ksplit; }

        // Stage A tile: 128 rows x 16 cols.
        {
            const int koff = (tid & 3) * 4;
#ifdef HAVE_ASYNC_LDS
            if (!bnScale) {
                // Raw copy: async DMA straight into LDS, no VGPR round-trip.
                for (int rr = 0; rr < 128; rr += 64) {
                    const int rowL = rr + (tid >> 2);
                    __builtin_amdgcn_global_load_async_to_lds_b128(
                        AS1(&src[(size_t)(rowBase + rowL) * ldS + kk + koff]),
                        AS3(&As[rowL][koff]), 0, 0);
                }
            } else
#endif
            {
                for (int rr = 0; rr < 128; rr += 64) {
                    const int rowL = rr + (tid >> 2);
                    const float4 v = *(const float4*)&src[(size_t)(rowBase + rowL) * ldS + kk + koff];
                    float x0 = v.x, x1 = v.y, x2 = v.z, x3 = v.w;
                    if (bnScale) {
                        const int kg = kb + koff;
                        x0 = x0 * bnScale[kg + 0] + bnShift[kg + 0];
                        x1 = x1 * bnScale[kg + 1] + bnShift[kg + 1];
                        x2 = x2 * bnScale[kg + 2] + bnShift[kg + 2];
                        x3 = x3 * bnScale[kg + 3] + bnShift[kg + 3];
                    }
                    As[rowL][koff + 0] = x0; As[rowL][koff + 1] = x1;
                    As[rowL][koff + 2] = x2; As[rowL][koff + 3] = x3;
                }
            }
        }
        // Stage B tile: 16 rows x 64 cols (256 threads exactly cover it).
        {
            const int kr   = tid >> 4;        // 0..15
            const int koff = (tid & 15) * 4;  // 0..60
#ifdef HAVE_ASYNC_LDS
            __builtin_amdgcn_global_load_async_to_lds_b128(
                AS1(&W[(size_t)(kb + kr) * 256 + colBase + koff]),
                AS3(&Bs[kr][koff]), 0, 0);
#else
            const float4 v = *(const float4*)&W[(size_t)(kb + kr) * 256 + colBase + koff];
            Bs[kr][koff + 0] = v.x; Bs[kr][koff + 1] = v.y;
            Bs[kr][koff + 2] = v.z; Bs[kr][koff + 3] = v.w;
#endif
        }
        // Prefetch next K-block of A while this one computes.
        if (kb + 16 < K) {
            const float* nsrc; int nldS; int nkk;
            if (kb + 16 < Ksplit) { nsrc = A;  nldS = ldA;  nkk = kb + 16; }
            else                  { nsrc = A2; nldS = ldA2; nkk = kb + 16 - Ksplit; }
            __builtin_prefetch(&nsrc[(size_t)(rowBase + (tid >> 1)) * nldS + nkk], 0, 3);
        }
#ifdef HAVE_ASYNC_LDS
        WAIT_ASYNC();   // per-wave: our async LDS writes have landed
#endif
        __syncthreads(); // cross-wave: everyone's tiles visible

        const int ar0 = waveM * 32 + lr;
        const int bc0 = waveN * 32 + lr;
#pragma unroll
        for (int ks = 0; ks < 16; ks += 4) {
            // A frag (16x4 f32): VGPR0 = K{0|2}, VGPR1 = K{1|3}; lanes 16-31 carry K+2.
            v2f a0 = *(const v2f*)&As[ar0][ks + 2 * hi];
            v2f a1 = *(const v2f*)&As[ar0 + 16][ks + 2 * hi];
            // B frags (4x16 f32), mirrored layout; two adjacent 16-col tiles.
            v2f b0, b1;
            b0.x = Bs[ks + 2 * hi][bc0];
            b0.y = Bs[ks + 2 * hi + 1][bc0];
            b1.x = Bs[ks + 2 * hi][bc0 + 16];
            b1.y = Bs[ks + 2 * hi + 1][bc0 + 16];
            acc00 = __builtin_amdgcn_wmma_f32_16x16x4_f32(false, a0, false, b0,
                                                          (short)0, acc00, false, false);
            acc01 = __builtin_amdgcn_wmma_f32_16x16x4_f32(false, a0, false, b1,
                                                          (short)0, acc01, false, false);
            acc10 = __builtin_amdgcn_wmma_f32_16x16x4_f32(false, a1, false, b0,
                                                          (short)0, acc10, false, false);
            acc11 = __builtin_amdgcn_wmma_f32_16x16x4_f32(false, a1, false, b1,
                                                          (short)0, acc11, false, false);
        }
        __syncthreads();
    }

    // Epilogue: bias + ReLU, D layout: VGPR v -> M = v + 8*hi, N = lr.
#pragma unroll
    for (int v = 0; v < 8; ++v) {
        const int row0 = rowBase + waveM * 32 + v + 8 * hi;
        const int row1 = row0 + 16;
        float r;
        r = acc00[v] + bias0; r = r > 0.f ? r : 0.f;
        Out[(size_t)row0 * 256 + colW0] = r;
        r = acc01[v] + bias1; r = r > 0.f ? r : 0.f;
        Out[(size_t)row0 * 256 + colW0 + 16] = r;
        r = acc10[v] + bias0; r = r > 0.f ? r : 0.f;
        Out[(size_t)row1 * 256 + colW0] = r;
        r = acc11[v] + bias1; r = r > 0.f ? r : 0.f;
        Out[(size_t)row1 * 256 + colW0 + 16] = r;
    }
}

// ---------------------------------------------------------------------------
// 4) Deterministic two-stage BN stats over 32768 rows x 256 channels.
// ---------------------------------------------------------------------------
__launch_bounds__(256)
__global__ void stats_partial_kernel(const float* __restrict__ H,
                                     float* __restrict__ psum,
                                     float* __restrict__ psumsq)
{
    const int c = threadIdx.x;
    const int rBase = blockIdx.x * 256;
    float s = 0.f, s2 = 0.f;
    for (int r = 0; r < 256; ++r) {
        const float v = H[(size_t)(rBase + r) * 256 + c];
        s += v; s2 += v * v;
    }
    psum[blockIdx.x * 256 + c]   = s;
    psumsq[blockIdx.x * 256 + c] = s2;
}

__launch_bounds__(256)
__global__ void stats_final_kernel(const float* __restrict__ psum,
                                   const float* __restrict__ psumsq,
                                   const float* __restrict__ g,
                                   const float* __restrict__ be,
                                   float* __restrict__ scale,
                                   float* __restrict__ shift)
{
    const int c = threadIdx.x;
    float s = 0.f, s2 = 0.f;
    for (int b = 0; b < 128; ++b) { s += psum[b * 256 + c]; s2 += psumsq[b * 256 + c]; }
    const float invN = 1.f / 32768.f;
    const float mu  = s * invN;
    const float var = s2 * invN - mu * mu;
    const float sc  = g[c] * rsqrtf(var + 1e-5f);
    scale[c] = sc;
    shift[c] = be[c] - mu * sc;
}

__launch_bounds__(256)
__global__ void apply_bn_kernel(const float* __restrict__ H,
                                const float* __restrict__ scale,
                                const float* __restrict__ shift,
                                float* __restrict__ out, int n)
{
    const int i = blockIdx.x * 256 + threadIdx.x;
    if (i < n) {
        const int c = i & 255;
        out[i] = H[i] * scale[c] + shift[c];
    }
}

// ---------------------------------------------------------------------------
// Orchestration
// ---------------------------------------------------------------------------
extern "C" void kernel_launch(void* const* d_in, const int* in_sizes, int n_in,
                              void* d_out, int out_size, void* d_ws, size_t ws_size,
                              hipStream_t stream)
{
    const float* x        = (const float*)d_in[0];
    const float* pos      = (const float*)d_in[1];
    // d_in[2] = batch (int64) — implicit from layout, unused
    const float* x_skip   = (const float*)d_in[3];
    const float* pos_skip = (const float*)d_in[4];
    // d_in[5] = batch_skip — unused
    const float* W1  = (const float*)d_in[6];
    const float* b1  = (const float*)d_in[7];
    const float* g1  = (const float*)d_in[8];
    const float* be1 = (const float*)d_in[9];
    const float* W2  = (const float*)d_in[10];
    const float* b2  = (const float*)d_in[11];
    const float* g2  = (const float*)d_in[12];
    const float* be2 = (const float*)d_in[13];

    char* ws = (char*)d_ws;
    float* Y      = (float*)(ws);                              // 32768*256 f32 (doubles as H2)
    int*   idx    = (int*)  (ws + 33554432);                   // 32768*3 ints
    float* wgt    = (float*)(ws + 33554432 + 393216);          // 32768*3 f32
    float* psum   = (float*)(ws + 33554432 + 786432);          // 128*256 f32
    float* psumsq = (float*)(ws + 33554432 + 786432 + 131072); // 128*256 f32
    float* scale1 = (float*)(ws + 33554432 + 786432 + 262144);
    float* shift1 = scale1 + 256;
    float* scale2 = scale1 + 512;
    float* shift2 = scale1 + 768;

    float* H1 = (float*)d_out;   // GEMM1 output lives in d_out until overwritten

    knn_kernel<<<128, 256, 0, stream>>>(pos, pos_skip, idx, wgt);
    interp_kernel<<<32768, 128, 0, stream>>>(x, idx, wgt, Y);
    // Layer 1: relu([Y | x_skip] @ W1 + b1) -> d_out (pre-BN)
    gemm_relu_kernel<<<dim3(256, 4), 256, 0, stream>>>(Y, 256, x_skip, 128,
                                                       256, 384, W1, b1,
                                                       nullptr, nullptr, H1);
    stats_partial_kernel<<<128, 256, 0, stream>>>(H1, psum, psumsq);
    stats_final_kernel<<<1, 256, 0, stream>>>(psum, psumsq, g1, be1, scale1, shift1);
    // Layer 2: relu(BN1(H1) @ W2 + b2) -> Y region (Y is dead), BN1 fused on A-load
    gemm_relu_kernel<<<dim3(256, 4), 256, 0, stream>>>(H1, 256, nullptr, 0,
                                                       256, 256, W2, b2,
                                                       scale1, shift1, Y);
    stats_partial_kernel<<<128, 256, 0, stream>>>(Y, psum, psumsq);
    stats_final_kernel<<<1, 256, 0, stream>>>(psum, psumsq, g2, be2, scale2, shift2);
    apply_bn_kernel<<<32768, 256, 0, stream>>>(Y, scale2, shift2, (float*)d_out, 32768 * 256);
}